// RLModel_70978629534115
// MI455X (gfx1250) — compile-verified
//
#include <hip/hip_runtime.h>
#include <hip/hip_bf16.h>
#include <math.h>

// Problem constants (match reference)
#define B_ 256
#define S_ 512
#define D_ 768
#define H_ 8
#define T_ 4
#define A_ (S_ + 1)          // 513
#define M_ (B_ * A_)         // 131328 rows of state_full

#define KC_ 32               // K-chunk per stage (8 WMMA k-steps of 4)
#define NCHUNK_ (D_ / KC_)   // 24

typedef __attribute__((ext_vector_type(2))) float v2f;
typedef __attribute__((ext_vector_type(8))) float v8f;

#if __has_builtin(__builtin_amdgcn_wmma_f32_16x16x4_f32)
#define HAVE_WMMA_F32 1
#else
#define HAVE_WMMA_F32 0
#endif

// ---------------------------------------------------------------------------
// Kernel 0: build padded weight matrix Wc(16 x 768) in workspace:
//   rows 0..7 = W_ft, row 8 = w_ext, rows 9..15 = 0
// Removes all exec-divergence from the GEMM B-fragment loads; 48 KB stays
// resident in WGP$/L2 for the whole grid.
// ---------------------------------------------------------------------------
__global__ void build_wc_kernel(const float* __restrict__ W_ft,
                                const float* __restrict__ w_ext,
                                float* __restrict__ Wc)
{
    int i = blockIdx.x * 256 + threadIdx.x;
    if (i < 16 * D_) {
        int n = i / D_;
        int d = i - n * D_;
        Wc[i] = (n < H_) ? W_ft[i] : ((n == H_) ? w_ext[d] : 0.0f);
    }
}

// ---------------------------------------------------------------------------
// Kernel 1: fused projection  proj(M,16) = state_full(M,768) x Wc^T(768,16)
// Bandwidth-bound (403 MB @ 23.3 TB/s ~ 17 us). f32 WMMA (16x16x4) keeps full
// reference precision. 8 waves/block, one 16-row tile per wave. A-tiles are
// DMA'd global->LDS with GLOBAL_LOAD_ASYNC_TO_LDS_B128 (ASYNCcnt), double-
// buffered so chunk i+1's DMA overlaps chunk i's WMMAs.
// ---------------------------------------------------------------------------
__launch_bounds__(256, 2)
__global__ void proj_kernel(const float* __restrict__ state,
                            const float* __restrict__ stop_emb,
                            const float* __restrict__ Wc,
                            float* __restrict__ proj)
{
    // 8 waves x 2 buffers x (16 rows x 32 K) f32 = 32 KB
    __shared__ float ldsA[8][2][16 * KC_];

    const int tid  = threadIdx.x;
    const int lane = tid & 31;
    const int wave = tid >> 5;
    const int tileRow = (blockIdx.x * 8 + wave) * 16;

    // Staging map: lane stages 4 float4s per chunk:
    //   row r = i*4 + (lane>>3), float4-col c4 = lane&7   (i = 0..3)
    const float* rowBase[4];
#pragma unroll
    for (int i = 0; i < 4; ++i) {
        int r   = i * 4 + (lane >> 3);
        int row = tileRow + r;
        int bb  = row / A_;
        int aa  = row - bb * A_;
        rowBase[i] = ((aa < S_) ? (state + ((size_t)bb * S_ + aa) * (size_t)D_)
                                : stop_emb)
                     + (lane & 7) * 4;
    }

    // LDS byte addresses (wave-relative) for the two buffers
    const unsigned ldsBuf0 = (unsigned)(uintptr_t)&ldsA[wave][0][0];
    const unsigned ldsBuf1 = (unsigned)(uintptr_t)&ldsA[wave][1][0];
    const unsigned ldsLaneOff = (unsigned)((lane >> 3) * (KC_ * 4) + (lane & 7) * 16);

    // B-fragment source (ISA layout: n = lane&15 column, k0 = (lane>>4)*2)
    const int n  = lane & 15;
    const int k0 = (lane >> 4) * 2;
    const float* bp = Wc + (size_t)n * D_ + k0;

    // ---- async-stage one K-chunk (4 x b128 per lane) into buffer ----
    auto stage = [&](int kb, unsigned bufBase) {
#pragma unroll
        for (int i = 0; i < 4; ++i) {
            unsigned ldst = bufBase + (unsigned)(i * 4 * KC_ * 4) + ldsLaneOff;
            const float* g = rowBase[i] + kb;
            asm volatile("global_load_async_to_lds_b128 %0, %1, off"
                         :: "v"(ldst), "v"(g) : "memory");
        }
    };

    v8f c = {};
    stage(0, ldsBuf0);

    for (int ch = 0; ch < NCHUNK_; ++ch) {
        const int buf = ch & 1;
        if (ch + 1 < NCHUNK_) {
            stage((ch + 1) * KC_, buf ? ldsBuf0 : ldsBuf1);
            asm volatile("s_wait_asynccnt 4" ::: "memory");   // chunk ch done
        } else {
            asm volatile("s_wait_asynccnt 0" ::: "memory");
        }

        const float* As = &ldsA[wave][buf][0];
#pragma unroll
        for (int k4 = 0; k4 < KC_ / 4; ++k4) {
            const float2 av = *(const float2*)&As[(lane & 15) * KC_ + k4 * 4 + k0];
            const float2 bv = *(const float2*)(bp + ch * KC_ + k4 * 4);
            v2f a, bf;
            a[0] = av.x; a[1] = av.y;
            bf[0] = bv.x; bf[1] = bv.y;
#if HAVE_WMMA_F32
            c = __builtin_amdgcn_wmma_f32_16x16x4_f32(false, a, false, bf,
                                                      (short)0, c, false, false);
#else
            // Scalar fallback with identical result layout
#pragma unroll
            for (int j = 0; j < 8; ++j) {
                int m = j + (lane >> 4) * 8;
                float acc = c[j];
#pragma unroll
                for (int k = 0; k < 4; ++k)
                    acc += As[m * KC_ + k4 * 4 + k] * bp[ch * KC_ + k4 * 4 + k - k0 + k];
                c[j] = acc;
            }
#endif
        }
    }

    // D layout: VGPR j -> (M = tileRow + j + (lane>>4)*8, N = lane&15)
#pragma unroll
    for (int j = 0; j < 8; ++j) {
        int row = tileRow + j + (lane >> 4) * 8;
        proj[(size_t)row * 16 + (lane & 15)] = c[j];
    }
}

// ---------------------------------------------------------------------------
// Kernel 2: actor sampling + critic GRU, one workgroup per batch element.
// ---------------------------------------------------------------------------
__device__ __forceinline__ unsigned pcg_hash(unsigned x)
{
    x = x * 747796405u + 2891336453u;
    unsigned w = ((x >> ((x >> 28u) + 4u)) ^ x) * 277803737u;
    return (w >> 22u) ^ w;
}

__launch_bounds__(256, 1)
__global__ void actor_critic_kernel(const float* __restrict__ proj,
                                    const float* __restrict__ mask,
                                    const float* __restrict__ W_ft,
                                    const float* __restrict__ init_hidden,
                                    const float* __restrict__ sos_emb,
                                    const float* __restrict__ W_ih,
                                    const float* __restrict__ W_hh,
                                    const float* __restrict__ b_ih,
                                    const float* __restrict__ b_hh,
                                    const float* __restrict__ W_v,
                                    const float* __restrict__ b_v,
                                    float* __restrict__ out)
{
    __shared__ float sft[A_ * H_];      // state_ft for this b (16.4 KB)
    __shared__ float probs[A_];
    __shared__ float zbuf[A_];          // logits / attention scores scratch
    __shared__ float maskb[A_];
    __shared__ float red[256];
    __shared__ int   redi[256];
    __shared__ float red8[256 * H_];    // ctx partial sums
    __shared__ float xs[(T_ + 1) * H_]; // GRU inputs (sos + ext_ft)
    __shared__ float hv[H_];
    __shared__ float ctxs[H_];
    __shared__ int   valid_s[T_];
    __shared__ int   active_s;
    __shared__ float max_steps_s;

    const int b   = blockIdx.x;
    const int tid = threadIdx.x;

    // ---- init: load state_ft, probs0 = sigmoid(proj[:,8]) * mask_full ----
    float msum = 0.0f;
    for (int a = tid; a < A_; a += 256) {
        const float* pr = proj + ((size_t)b * A_ + a) * 16;
        float4 p0 = *(const float4*)pr;
        float4 p1 = *(const float4*)(pr + 4);
        float* s8 = &sft[a * H_];
        s8[0] = p0.x; s8[1] = p0.y; s8[2] = p0.z; s8[3] = p0.w;
        s8[4] = p1.x; s8[5] = p1.y; s8[6] = p1.z; s8[7] = p1.w;
        float mk = (a < S_) ? mask[(size_t)b * S_ + a] : 1.0f;
        maskb[a] = mk;
        float pe = pr[8];
        probs[a] = (1.0f / (1.0f + __expf(-pe))) * mk;
        msum += mk;
    }
    red[tid] = msum;
    __syncthreads();
    for (int off = 128; off > 0; off >>= 1) {
        if (tid < off) red[tid] += red[tid + off];
        __syncthreads();
    }
    if (tid == 0) {
        max_steps_s = fminf((float)T_, red[0]);
        active_s = 1;
    }
    if (tid < H_) {
        float s = 0.0f;
        for (int d = 0; d < D_; ++d) s += sos_emb[d] * W_ft[tid * D_ + d];
        xs[tid] = s;                    // SOS embedding @ W_ft^T
        hv[tid] = init_hidden[tid];
    }
    __syncthreads();

    // ---------------- actor: T autoregressive sampling steps ----------------
    for (int t = 0; t < T_; ++t) {
        // pass 1: logits z = log(p) - log1p(-p), block max
        float zmax = -3.4e38f;
        for (int a = tid; a < A_; a += 256) {
            float p = fmaxf(probs[a], 1e-6f);
            float z = __logf(p) - log1pf(-p);
            zbuf[a] = z;
            zmax = fmaxf(zmax, z);
        }
        red[tid] = zmax;
        __syncthreads();
        for (int off = 128; off > 0; off >>= 1) {
            if (tid < off) red[tid] = fmaxf(red[tid], red[tid + off]);
            __syncthreads();
        }
        zmax = red[0];
        __syncthreads();

        // pass 2: sum-exp + Gumbel-max categorical sample
        float se = 0.0f, gbest = -3.4e38f;
        int gidx = 0;
        for (int a = tid; a < A_; a += 256) {
            float z = zbuf[a];
            se += __expf(z - zmax);
            unsigned h = pcg_hash(((unsigned)(t * B_ + b)) * 2654435761u
                                  ^ ((unsigned)a * 0x9E3779B9u));
            float u = ((float)h + 0.5f) * (1.0f / 4294967296.0f);
            float g = z - __logf(-__logf(u));      // Gumbel noise
            if (g > gbest) { gbest = g; gidx = a; }
        }
        red[tid] = se;
        __syncthreads();
        for (int off = 128; off > 0; off >>= 1) {
            if (tid < off) red[tid] += red[tid + off];
            __syncthreads();
        }
        float sumE = red[0];
        __syncthreads();
        red[tid] = gbest; redi[tid] = gidx;
        __syncthreads();
        for (int off = 128; off > 0; off >>= 1) {
            if (tid < off && red[tid + off] > red[tid]) {
                red[tid] = red[tid + off]; redi[tid] = redi[tid + off];
            }
            __syncthreads();
        }
        int action = redi[0];
        float lse = zmax + __logf(sumE);
        __syncthreads();

        // pass 3: entropy
        float es = 0.0f;
        for (int a = tid; a < A_; a += 256) {
            float lp = zbuf[a] - lse;
            es += __expf(lp) * lp;
        }
        red[tid] = es;
        __syncthreads();
        for (int off = 128; off > 0; off >>= 1) {
            if (tid < off) red[tid] += red[tid + off];
            __syncthreads();
        }
        if (tid < H_) xs[(t + 1) * H_ + tid] = sft[action * H_ + tid]; // ext_ft
        if (tid == 0) {
            int valid = active_s;
            float lp  = zbuf[action] - lse;
            float ent = -red[0];
            out[((size_t)b * T_ + t) * 3 + 0] = valid ? lp : 0.0f;
            out[((size_t)b * T_ + t) * 3 + 1] = valid ? ent : 0.0f;
            out[(size_t)B_ * T_ * 3 + (size_t)b * T_ + t] = (float)action;
            valid_s[t] = valid;
            probs[action] = 0.0f;                  // mask extracted sentence
            active_s = valid && (action < A_ - 1)
                             && ((float)(t + 1) < max_steps_s);
        }
        __syncthreads();
    }

    // ---------------- critic: T GRU+attention steps ----------------
    for (int t = 0; t < T_; ++t) {
        // attention scores
        float amax = -3.4e38f;
        for (int a = tid; a < A_; a += 256) {
            const float* s8 = &sft[a * H_];
            float att = 0.0f;
#pragma unroll
            for (int h = 0; h < H_; ++h) att += hv[h] * s8[h];
            att = (maskb[a] > 0.0f) ? att : -1e6f;
            zbuf[a] = att;
            amax = fmaxf(amax, att);
        }
        red[tid] = amax;
        __syncthreads();
        for (int off = 128; off > 0; off >>= 1) {
            if (tid < off) red[tid] = fmaxf(red[tid], red[tid + off]);
            __syncthreads();
        }
        amax = red[0];
        __syncthreads();

        float se = 0.0f;
        float ctxp[H_];
#pragma unroll
        for (int h = 0; h < H_; ++h) ctxp[h] = 0.0f;
        for (int a = tid; a < A_; a += 256) {
            float e = __expf(zbuf[a] - amax);
            se += e;
            const float* s8 = &sft[a * H_];
#pragma unroll
            for (int h = 0; h < H_; ++h) ctxp[h] += e * s8[h];
        }
        red[tid] = se;
        __syncthreads();
        for (int off = 128; off > 0; off >>= 1) {
            if (tid < off) red[tid] += red[tid + off];
            __syncthreads();
        }
        float sumE = red[0];
#pragma unroll
        for (int h = 0; h < H_; ++h) red8[h * 256 + tid] = ctxp[h];
        __syncthreads();
        if (tid < H_) {
            float s = 0.0f;
            for (int i = 0; i < 256; ++i) s += red8[tid * 256 + i];
            ctxs[tid] = s / sumE;
        }
        __syncthreads();

        if (tid == 0) {
            // GRU cell (gate order r,z,n) + value head
            float x[2 * H_];
#pragma unroll
            for (int j = 0; j < H_; ++j) { x[j] = xs[t * H_ + j]; x[H_ + j] = ctxs[j]; }
            float gi[3 * H_], gh[3 * H_];
            for (int g = 0; g < 3 * H_; ++g) {
                float si = b_ih[g], sh = b_hh[g];
                for (int j = 0; j < 2 * H_; ++j) si += W_ih[g * 2 * H_ + j] * x[j];
                for (int j = 0; j < H_; ++j)     sh += W_hh[g * H_ + j] * hv[j];
                gi[g] = si; gh[g] = sh;
            }
            float hn[H_];
            float val = b_v[0];
#pragma unroll
            for (int j = 0; j < H_; ++j) {
                float r  = 1.0f / (1.0f + __expf(-(gi[j] + gh[j])));
                float zz = 1.0f / (1.0f + __expf(-(gi[H_ + j] + gh[H_ + j])));
                float nn = tanhf(gi[2 * H_ + j] + r * gh[2 * H_ + j]);
                hn[j] = (1.0f - zz) * nn + zz * hv[j];
            }
            for (int j = 0; j < H_; ++j) val += W_v[j] * hn[j];
            int valid = valid_s[t];
            out[((size_t)b * T_ + t) * 3 + 2] = valid ? val : 0.0f;
            if (valid) {
#pragma unroll
                for (int j = 0; j < H_; ++j) hv[j] = hn[j];
            }
        }
        __syncthreads();
    }
}

// ---------------------------------------------------------------------------
extern "C" void kernel_launch(void* const* d_in, const int* in_sizes, int n_in,
                              void* d_out, int out_size, void* d_ws, size_t ws_size,
                              hipStream_t stream)
{
    (void)in_sizes; (void)n_in; (void)out_size; (void)ws_size;

    const float* state       = (const float*)d_in[0];
    const float* mask        = (const float*)d_in[1];
    const float* w_ext       = (const float*)d_in[2];
    const float* stop_emb    = (const float*)d_in[3];
    const float* W_ft        = (const float*)d_in[4];
    const float* init_hidden = (const float*)d_in[5];
    const float* sos_emb     = (const float*)d_in[6];
    const float* W_ih        = (const float*)d_in[7];
    const float* W_hh        = (const float*)d_in[8];
    const float* b_ih        = (const float*)d_in[9];
    const float* b_hh        = (const float*)d_in[10];
    const float* W_v         = (const float*)d_in[11];
    const float* b_v         = (const float*)d_in[12];

    float* proj = (float*)d_ws;                       // M_ x 16 f32 (~8.4 MB)
    float* Wc   = (float*)d_ws + (size_t)M_ * 16;     // 16 x 768 f32 (48 KB)
    float* out  = (float*)d_out;                      // (B,T,3) then (B,T)

    build_wc_kernel<<<(16 * D_ + 255) / 256, 256, 0, stream>>>(W_ft, w_ext, Wc);
    proj_kernel<<<M_ / 128, 256, 0, stream>>>(state, stop_emb, Wc, proj);
    actor_critic_kernel<<<B_, 256, 0, stream>>>(proj, mask, W_ft, init_hidden,
                                                sos_emb, W_ih, W_hh, b_ih, b_hh,
                                                W_v, b_v, out);
}